// LightGCN_40553081209509
// MI455X (gfx1250) — compile-verified
//
#include <hip/hip_runtime.h>
#include <cstdint>

// LightGCN: acc = emb0; emb = SpMM(A, emb) x3 accumulating; out = acc/4.
// Memory/atomic bound; embedding ping-pong buffers (2 x 76.8MB) fit in the
// MI455X 192MB L2 and are the hot set (2.4GB of gather hits per layer), so
// the once-per-pass edge stream (115MB/layer) is loaded NON-TEMPORAL.
// CDNA5 paths used:
//   - GLOBAL_LOAD_ASYNC_TO_LDS_B128 th:TH_LOAD_NT double-buffered edge
//     staging, ordered with s_wait_asynccnt (ASYNCcnt)
//   - global_prefetch_b8 of the next edge's embedding row
//   - wave32-aware 16-lane edge groups, b128 vector loads, native
//     no-return global_atomic_add_f32 scatters

#define EMBED_DIM   64
#define TILE_EDGES  1024
#define SPMM_BLOCK  256

// ---------- CDNA5 async-to-LDS helpers (inline asm, per CDNA5 ISA) ------
__device__ __forceinline__ uint32_t lds_lo32(const void* p) {
  // Generic (flat) pointers to LDS carry the LDS byte offset in the low 32
  // bits (ISA 10.2 aperture table).
  return (uint32_t)(uintptr_t)p;
}

__device__ __forceinline__ void async_load_b128_nt(uint32_t lds_off, const void* gaddr) {
  // Non-temporal: edge stream is single-use per pass; keep L2 for embeddings.
  asm volatile("global_load_async_to_lds_b128 %0, %1, off th:TH_LOAD_NT"
               :: "v"(lds_off), "v"(gaddr)
               : "memory");
}

__device__ __forceinline__ void wait_async_le3() {
  asm volatile("s_wait_asynccnt 3" ::: "memory");
}
__device__ __forceinline__ void wait_async_0() {
  asm volatile("s_wait_asynccnt 0" ::: "memory");
}

// ---------------------------- SpMM (COO scatter) ---------------------------
__global__ __launch_bounds__(SPMM_BLOCK)
void lightgcn_spmm_kernel(const int* __restrict__ rows,
                          const int* __restrict__ cols,
                          const float* __restrict__ vals,
                          const float* __restrict__ x,
                          float* __restrict__ y,
                          int nnz) {
  __shared__ int   sh_rows[2][TILE_EDGES];
  __shared__ int   sh_cols[2][TILE_EDGES];
  __shared__ float sh_vals[2][TILE_EDGES];

  const int tid    = threadIdx.x;
  const int ntiles = (nnz + TILE_EDGES - 1) / TILE_EDGES;

  // Stage one tile of edge data into LDS buffer `buf`.
  // Full tile: 3 async b128 per thread (256 thr * 16B = 4KB = one array).
  // Partial tail tile (cannot happen here: 9.6M % 1024 == 0, but be safe):
  // plain guarded LDS stores. Returns 1 iff async path was used.
  auto stage = [&](int tile, int buf) -> int {
    const int base = tile * TILE_EDGES;
    const int cnt  = min(TILE_EDGES, nnz - base);
    if (cnt == TILE_EDGES) {
      const uint32_t boff = (uint32_t)tid * 16u;
      async_load_b128_nt(lds_lo32(&sh_rows[buf][0]) + boff, (const char*)(rows + base) + boff);
      async_load_b128_nt(lds_lo32(&sh_cols[buf][0]) + boff, (const char*)(cols + base) + boff);
      async_load_b128_nt(lds_lo32(&sh_vals[buf][0]) + boff, (const char*)(vals + base) + boff);
      return 1;
    }
    for (int i = tid; i < cnt; i += SPMM_BLOCK) {
      sh_rows[buf][i] = rows[base + i];
      sh_cols[buf][i] = cols[base + i];
      sh_vals[buf][i] = vals[base + i];
    }
    return 0;
  };

  int tile = blockIdx.x;
  if (tile >= ntiles) return;

  const int stride = gridDim.x;
  int buf = 0;
  (void)stage(tile, buf);  // prologue: first tile in flight

  const int lane16 = tid & 15;        // lane within 16-wide edge group
  const int doff   = lane16 << 2;     // 4 dims per lane (float4)
  const int group  = tid >> 4;        // 16 edge groups per 256-thread block

  for (; tile < ntiles; tile += stride) {
    const int ntile = tile + stride;
    int nxt_async = 0;
    if (ntile < ntiles) nxt_async = stage(ntile, buf ^ 1);  // prefetch next tile

    // Wait until only the *next* tile's 3 async loads remain outstanding
    // (async loads complete in order), then make LDS visible block-wide.
    if (nxt_async) wait_async_le3();
    else           wait_async_0();
    __syncthreads();

    const int base = tile * TILE_EDGES;
    const int cnt  = min(TILE_EDGES, nnz - base);
    for (int i = group; i < cnt; i += (SPMM_BLOCK / 16)) {
      const int   c = sh_cols[buf][i];
      const int   r = sh_rows[buf][i];
      const float v = sh_vals[buf][i];

      // Prefetch next iteration's gather row (global_prefetch_b8).
      const int inext = i + (SPMM_BLOCK / 16);
      if (inext < cnt) {
        const int cn = sh_cols[buf][inext];
        __builtin_prefetch(x + cn * EMBED_DIM + doff, 0, 3);
      }

      // Coalesced 256B gather per edge group (global_load_b128 per lane).
      const float4 xv = *reinterpret_cast<const float4*>(x + c * EMBED_DIM + doff);
      float* yp = y + r * EMBED_DIM + doff;
      unsafeAtomicAdd(yp + 0, xv.x * v);
      unsafeAtomicAdd(yp + 1, xv.y * v);
      unsafeAtomicAdd(yp + 2, xv.z * v);
      unsafeAtomicAdd(yp + 3, xv.w * v);
    }
    __syncthreads();  // buf may be overwritten by next iteration's prefetch
    buf ^= 1;
  }
}

// ------------------------- elementwise helper kernels ----------------------
__global__ void lightgcn_init_kernel(const float4* __restrict__ user4,
                                     const float4* __restrict__ item4,
                                     float4* __restrict__ curr4,
                                     float4* __restrict__ next4,
                                     float4* __restrict__ out4,
                                     int users4, int total4) {
  const int i = blockIdx.x * blockDim.x + threadIdx.x;
  if (i >= total4) return;
  const float4 s = (i < users4) ? user4[i] : item4[i - users4];
  curr4[i] = s;                                   // layer-0 embeddings
  out4[i]  = s;                                   // acc = emb0
  next4[i] = make_float4(0.f, 0.f, 0.f, 0.f);     // zero scatter target
}

__global__ void lightgcn_accum_zero_kernel(float4* __restrict__ out4,
                                           const float4* __restrict__ add4,
                                           float4* __restrict__ zero4,
                                           int total4) {
  const int i = blockIdx.x * blockDim.x + threadIdx.x;
  if (i >= total4) return;
  float4 o = out4[i];
  const float4 a = add4[i];
  o.x += a.x; o.y += a.y; o.z += a.z; o.w += a.w;
  out4[i]  = o;
  zero4[i] = make_float4(0.f, 0.f, 0.f, 0.f);     // prep next scatter target
}

__global__ void lightgcn_final_kernel(float4* __restrict__ out4,
                                      const float4* __restrict__ add4,
                                      int total4) {
  const int i = blockIdx.x * blockDim.x + threadIdx.x;
  if (i >= total4) return;
  float4 o = out4[i];
  const float4 a = add4[i];
  o.x = (o.x + a.x) * 0.25f;
  o.y = (o.y + a.y) * 0.25f;
  o.z = (o.z + a.z) * 0.25f;
  o.w = (o.w + a.w) * 0.25f;
  out4[i] = o;
}

// --------------------------------- launch ----------------------------------
extern "C" void kernel_launch(void* const* d_in, const int* in_sizes, int n_in,
                              void* d_out, int out_size, void* d_ws, size_t ws_size,
                              hipStream_t stream) {
  const int*   rows = (const int*)d_in[0];
  const int*   cols = (const int*)d_in[1];
  const float* vals = (const float*)d_in[2];
  const float* user = (const float*)d_in[3];
  const float* item = (const float*)d_in[4];

  const int nnz     = in_sizes[0];
  const int n_users = in_sizes[3] / EMBED_DIM;
  const int n_items = in_sizes[4] / EMBED_DIM;
  const int n_nodes = n_users + n_items;

  float* out  = (float*)d_out;
  float* bufA = (float*)d_ws;                                  // ping
  float* bufB = bufA + (size_t)n_nodes * EMBED_DIM;            // pong

  const int total4 = n_nodes * (EMBED_DIM / 4);
  const int users4 = n_users * (EMBED_DIM / 4);
  const int eb     = (total4 + 255) / 256;

  const int ntiles = (nnz + TILE_EDGES - 1) / TILE_EDGES;
  const int sg     = ntiles < 2048 ? ntiles : 2048;  // >=2 tiles/block => double-buffer pays off

  // acc = emb0; bufA = emb0; bufB = 0
  lightgcn_init_kernel<<<eb, 256, 0, stream>>>(
      (const float4*)user, (const float4*)item,
      (float4*)bufA, (float4*)bufB, (float4*)out, users4, total4);

  // Layer 1: e1 = A*emb0
  lightgcn_spmm_kernel<<<sg, SPMM_BLOCK, 0, stream>>>(rows, cols, vals, bufA, bufB, nnz);
  lightgcn_accum_zero_kernel<<<eb, 256, 0, stream>>>(
      (float4*)out, (const float4*)bufB, (float4*)bufA, total4);

  // Layer 2: e2 = A*e1
  lightgcn_spmm_kernel<<<sg, SPMM_BLOCK, 0, stream>>>(rows, cols, vals, bufB, bufA, nnz);
  lightgcn_accum_zero_kernel<<<eb, 256, 0, stream>>>(
      (float4*)out, (const float4*)bufA, (float4*)bufB, total4);

  // Layer 3: e3 = A*e2 ; out = (acc + e3) / 4
  lightgcn_spmm_kernel<<<sg, SPMM_BLOCK, 0, stream>>>(rows, cols, vals, bufA, bufB, nnz);
  lightgcn_final_kernel<<<eb, 256, 0, stream>>>(
      (float4*)out, (const float4*)bufB, total4);
}